// NOAD_35519379537928
// MI455X (gfx1250) — compile-verified
//
#include <hip/hip_runtime.h>

// Problem constants (match reference setup_inputs)
#define B_   16
#define C1_  512
#define H1_  64
#define C2_  1024
#define H2_  32
#define OUT_ 256
#define P_   2048
#define S1_  (H1_ * H1_)   // 4096 pixels at 64x64
#define S2_  (H2_ * H2_)   // 1024 pixels at 32x32
#define CT_  (C1_ + C2_)   // 1536 = proj_w row stride

typedef float v2f __attribute__((ext_vector_type(2)));
typedef float v8f __attribute__((ext_vector_type(8)));

// D = A(16x4 f32) * B(4x16 f32) + C(16x16 f32), full fp32 matrix core.
static __device__ __forceinline__ v8f wmma4(v2f a, v2f b, v8f c) {
  return __builtin_amdgcn_wmma_f32_16x16x4_f32(false, a, false, b, (short)0, c,
                                               false, false);
}

// swap lanes L <-> L+16 (ds_swizzle group-of-32: and=0x1f, or=0, xor=0x10)
static __device__ __forceinline__ float swap16f(float v) {
  return __int_as_float(__builtin_amdgcn_ds_swizzle(__float_as_int(v), 0x401f));
}

// keep 3 smallest, t0 <= t1 <= t2
static __device__ __forceinline__ void ins3(float v, float& t0, float& t1, float& t2) {
  if (v < t2) {
    if (v < t1) {
      if (v < t0) { t2 = t1; t1 = t0; t0 = v; }
      else        { t2 = t1; t1 = v; }
    } else {
      t2 = v;
    }
  }
}

// ---------------------------------------------------------------------------
// cnorm[p] = ||centroids[p]||^2
__global__ void k_cnorm(const float* __restrict__ C, float* __restrict__ CN) {
  int p = blockIdx.x * blockDim.x + threadIdx.x;
  if (p >= P_) return;
  const float* r = C + (size_t)p * OUT_;
  float s = 0.f;
  for (int o = 0; o < OUT_; ++o) s += r[o] * r[o];
  CN[p] = s;
}

// ---------------------------------------------------------------------------
// E[b][o][s] = sum_c W[o][coff+c] * X[b][c][s]   (1x1 conv before pooling;
// exact because pooling/resize commute with a channel-wise linear map).
// One wave computes a 32(M=o) x 64(N=s) tile: 8 fp32-WMMA accumulators,
// per K-step 2 x b64 A-loads + 8 x b32 B-loads feed 8 WMMAs.
__global__ __launch_bounds__(256) void k_proj(const float* __restrict__ X,
                                              const float* __restrict__ W,
                                              float* __restrict__ E,
                                              int C, int S, int coff) {
  const int lane = threadIdx.x & 31;
  const int wid  = (blockIdx.x * blockDim.x + threadIdx.x) >> 5;
  const int ntiles = S >> 6;
  const int b   = wid / (8 * ntiles);
  const int rr  = wid % (8 * ntiles);
  const int ot  = rr / ntiles;      // 0..7, o-base = ot*32
  const int nt  = rr % ntiles;
  const int half = lane >> 4;       // K-pair selector (ISA f32 A/B layout)
  const int l16  = lane & 15;       // M (for A) / N (for B,D)
  const int n0   = nt << 6;

  const float* __restrict__ wrow0 = W + (size_t)(ot * 32 + l16) * CT_ + coff + 2 * half;
  const float* __restrict__ wrow1 = wrow0 + (size_t)16 * CT_;
  const float* __restrict__ xb    = X + (size_t)b * C * S + n0 + l16;

  v8f a00 = {}, a01 = {}, a02 = {}, a03 = {};
  v8f a10 = {}, a11 = {}, a12 = {}, a13 = {};
  for (int k = 0; k < C; k += 4) {
    v2f af0; af0.x = wrow0[k]; af0.y = wrow0[k + 1];
    v2f af1; af1.x = wrow1[k]; af1.y = wrow1[k + 1];
    const float* xr = xb + (size_t)(k + 2 * half) * S;
    v2f b0; b0.x = xr[0];  b0.y = xr[S + 0];
    v2f b1; b1.x = xr[16]; b1.y = xr[S + 16];
    v2f b2; b2.x = xr[32]; b2.y = xr[S + 32];
    v2f b3; b3.x = xr[48]; b3.y = xr[S + 48];
    a00 = wmma4(af0, b0, a00);
    a01 = wmma4(af0, b1, a01);
    a02 = wmma4(af0, b2, a02);
    a03 = wmma4(af0, b3, a03);
    a10 = wmma4(af1, b0, a10);
    a11 = wmma4(af1, b1, a11);
    a12 = wmma4(af1, b2, a12);
    a13 = wmma4(af1, b3, a13);
  }
  float* __restrict__ eb = E + (size_t)b * OUT_ * S + n0 + l16;
#pragma unroll
  for (int r = 0; r < 8; ++r) {
    float* row0 = eb + (size_t)(ot * 32 + r + 8 * half) * S;
    float* row1 = row0 + (size_t)16 * S;
    row0[0] = a00[r]; row0[16] = a01[r]; row0[32] = a02[r]; row0[48] = a03[r];
    row1[0] = a10[r]; row1[16] = a11[r]; row1[32] = a12[r]; row1[48] = a13[r];
  }
}

// ---------------------------------------------------------------------------
// 3x3 avg pool (count_include_pad -> always /9) of e2 at 32x32
__global__ void k_pool2(const float* __restrict__ E2, float* __restrict__ P2) {
  int idx = blockIdx.x * blockDim.x + threadIdx.x;
  if (idx >= B_ * OUT_ * S2_) return;
  int w  = idx & (H2_ - 1);
  int h  = (idx >> 5) & (H2_ - 1);
  int bo = idx >> 10;
  const float* r = E2 + (size_t)bo * S2_;
  float s = 0.f;
#pragma unroll
  for (int dh = -1; dh <= 1; ++dh) {
    int hh = h + dh; if ((unsigned)hh >= H2_) continue;
#pragma unroll
    for (int dw = -1; dw <= 1; ++dw) {
      int ww = w + dw; if ((unsigned)ww >= H2_) continue;
      s += r[hh * H2_ + ww];
    }
  }
  P2[idx] = s * (1.f / 9.f);
}

// ---------------------------------------------------------------------------
// EMB[b][o][n] = pool3(e1)[b][o][n] + bilinear_up2(P2)[b][o][n] + bias[o]
// jax.image.resize bilinear, half-pixel centers, edge weights renormalized:
//   out[2k]   = (k>0)  ? 0.25*in[k-1] + 0.75*in[k]   : in[0]
//   out[2k+1] = (k<31) ? 0.75*in[k]   + 0.25*in[k+1] : in[31]
__global__ void k_emb(const float* __restrict__ E1, const float* __restrict__ P2,
                      const float* __restrict__ bias, float* __restrict__ EMB) {
  int idx = blockIdx.x * blockDim.x + threadIdx.x;     // B*OUT*S1 threads
  int n  = idx & (S1_ - 1);
  int bo = idx >> 12;
  int o  = bo & (OUT_ - 1);
  int h = n >> 6, w = n & 63;

  const float* r1 = E1 + (size_t)bo * S1_;
  float s = 0.f;
#pragma unroll
  for (int dh = -1; dh <= 1; ++dh) {
    int hh = h + dh; if ((unsigned)hh >= H1_) continue;
#pragma unroll
    for (int dw = -1; dw <= 1; ++dw) {
      int ww = w + dw; if ((unsigned)ww >= H1_) continue;
      s += r1[hh * H1_ + ww];
    }
  }
  s *= (1.f / 9.f);

  int kh = h >> 1; int h0, h1; float wh0, wh1;
  if (h & 1) { if (kh < H2_ - 1) { h0 = kh; wh0 = 0.75f; h1 = kh + 1; wh1 = 0.25f; }
               else              { h0 = h1 = H2_ - 1; wh0 = 1.f; wh1 = 0.f; } }
  else       { if (kh > 0)       { h0 = kh - 1; wh0 = 0.25f; h1 = kh; wh1 = 0.75f; }
               else              { h0 = h1 = 0; wh0 = 1.f; wh1 = 0.f; } }
  int kw = w >> 1; int w0, w1; float ww0, ww1;
  if (w & 1) { if (kw < H2_ - 1) { w0 = kw; ww0 = 0.75f; w1 = kw + 1; ww1 = 0.25f; }
               else              { w0 = w1 = H2_ - 1; ww0 = 1.f; ww1 = 0.f; } }
  else       { if (kw > 0)       { w0 = kw - 1; ww0 = 0.25f; w1 = kw; ww1 = 0.75f; }
               else              { w0 = w1 = 0; ww0 = 1.f; ww1 = 0.f; } }

  const float* r2 = P2 + (size_t)bo * S2_;
  float bil = wh0 * (ww0 * r2[h0 * H2_ + w0] + ww1 * r2[h0 * H2_ + w1]) +
              wh1 * (ww0 * r2[h1 * H2_ + w0] + ww1 * r2[h1 * H2_ + w1]);

  EMB[idx] = s + bil + bias[o];
}

// ---------------------------------------------------------------------------
// feats[b][n] = sum_o EMB[b][o][n]^2  (coalesced: threads contiguous in n)
__global__ void k_feats(const float* __restrict__ EMB, float* __restrict__ F) {
  int idx = blockIdx.x * blockDim.x + threadIdx.x;     // B*S1 threads
  int n = idx & (S1_ - 1), b = idx >> 12;
  const float* e = EMB + (size_t)b * OUT_ * S1_ + n;
  float s = 0.f;
  for (int o = 0; o < OUT_; ++o) {
    float v = e[(size_t)o * S1_];
    s += v * v;
  }
  F[idx] = s;
}

// ---------------------------------------------------------------------------
// Distance GEMM + top-3 + softmin score. M = 64 centroid rows, N = two 16-pixel
// tiles (32 pixels per wave), K = 256 embed dims. In the f32 D layout lane L
// owns pixel n0 + (L&15) (and n0+16+(L&15) for the second tile), so top-3 of
// dist^2 is a per-lane running update; lanes L and L+16 cover the two row
// halves of p and are merged once at the end with ds_swizzle SWAPX16.
__global__ __launch_bounds__(256) void k_dist(const float* __restrict__ EMB,
                                              const float* __restrict__ CENT,
                                              const float* __restrict__ F,
                                              const float* __restrict__ CN,
                                              float* __restrict__ O) {
  const int lane = threadIdx.x & 31;
  const int wid  = (blockIdx.x * blockDim.x + threadIdx.x) >> 5;  // 0..2047
  const int half = lane >> 4, l16 = lane & 15;
  const int b = wid >> 7, nt = wid & 127;   // 128 tiles of 32 pixels per batch
  const int n0 = nt * 32;

  const float* __restrict__ eb0 = EMB + (size_t)b * OUT_ * S1_ + n0 + l16;
  const float* __restrict__ eb1 = eb0 + 16;
  const float fe0 = F[b * S1_ + n0 + l16];
  const float fe1 = F[b * S1_ + n0 + 16 + l16];
  float t0 = 3.4e38f, t1 = 3.4e38f, t2 = 3.4e38f;
  float s0 = 3.4e38f, s1 = 3.4e38f, s2 = 3.4e38f;

  for (int p0 = 0; p0 < P_; p0 += 64) {
    v8f d00 = {}, d01 = {}, d02 = {}, d03 = {};
    v8f d10 = {}, d11 = {}, d12 = {}, d13 = {};
    const float* __restrict__ ar = CENT + (size_t)(p0 + l16) * OUT_ + 2 * half;
    for (int k = 0; k < OUT_; k += 4) {
      const float* er0 = eb0 + (size_t)(k + 2 * half) * S1_;
      const float* er1 = eb1 + (size_t)(k + 2 * half) * S1_;
      v2f bb0; bb0.x = er0[0]; bb0.y = er0[S1_];
      v2f bb1; bb1.x = er1[0]; bb1.y = er1[S1_];
      v2f a0, a1, a2, a3;
      a0.x = ar[k];             a0.y = ar[k + 1];
      a1.x = ar[k + 16 * OUT_]; a1.y = ar[k + 16 * OUT_ + 1];
      a2.x = ar[k + 32 * OUT_]; a2.y = ar[k + 32 * OUT_ + 1];
      a3.x = ar[k + 48 * OUT_]; a3.y = ar[k + 48 * OUT_ + 1];
      d00 = wmma4(a0, bb0, d00);
      d01 = wmma4(a1, bb0, d01);
      d02 = wmma4(a2, bb0, d02);
      d03 = wmma4(a3, bb0, d03);
      d10 = wmma4(a0, bb1, d10);
      d11 = wmma4(a1, bb1, d11);
      d12 = wmma4(a2, bb1, d12);
      d13 = wmma4(a3, bb1, d13);
    }
#pragma unroll
    for (int r = 0; r < 8; ++r) {
      int pr = p0 + r + 8 * half;
      float c0 = CN[pr + 0], c16 = CN[pr + 16], c32 = CN[pr + 32], c48 = CN[pr + 48];
      ins3(fe0 + c0  - 2.f * d00[r], t0, t1, t2);
      ins3(fe0 + c16 - 2.f * d01[r], t0, t1, t2);
      ins3(fe0 + c32 - 2.f * d02[r], t0, t1, t2);
      ins3(fe0 + c48 - 2.f * d03[r], t0, t1, t2);
      ins3(fe1 + c0  - 2.f * d10[r], s0, s1, s2);
      ins3(fe1 + c16 - 2.f * d11[r], s0, s1, s2);
      ins3(fe1 + c32 - 2.f * d12[r], s0, s1, s2);
      ins3(fe1 + c48 - 2.f * d13[r], s0, s1, s2);
    }
  }

  // merge the two half-wave top-3 sets for each pixel
  float u0 = swap16f(t0), u1 = swap16f(t1), u2 = swap16f(t2);
  ins3(u0, t0, t1, t2); ins3(u1, t0, t1, t2); ins3(u2, t0, t1, t2);
  float v0 = swap16f(s0), v1 = swap16f(s1), v2 = swap16f(s2);
  ins3(v0, s0, s1, s2); ins3(v1, s0, s1, s2); ins3(v2, s0, s1, s2);

  float ta = sqrtf(t0), tb = sqrtf(t1), tc = sqrtf(t2);
  float sa = sqrtf(s0), sb = sqrtf(s1), sc = sqrtf(s2);
  float wt = 1.f / (1.f + __expf(ta - tb) + __expf(ta - tc));
  float ws = 1.f / (1.f + __expf(sa - sb) + __expf(sa - sc));
  if (half == 0) {
    O[b * S1_ + n0 + l16]      = wt * ta;
    O[b * S1_ + n0 + 16 + l16] = ws * sa;
  }
}

// ---------------------------------------------------------------------------
extern "C" void kernel_launch(void* const* d_in, const int* in_sizes, int n_in,
                              void* d_out, int out_size, void* d_ws, size_t ws_size,
                              hipStream_t stream) {
  (void)in_sizes; (void)n_in; (void)out_size; (void)ws_size;
  const float* p1 = (const float*)d_in[0];  // [16,512,64,64]
  const float* p2 = (const float*)d_in[1];  // [16,1024,32,32]
  const float* pw = (const float*)d_in[2];  // [256,1536]
  const float* pb = (const float*)d_in[3];  // [256]
  const float* ce = (const float*)d_in[4];  // [2048,256]  (d_in[5] = K = 3)
  float* out = (float*)d_out;               // [16,64,64]
  float* ws  = (float*)d_ws;

  // workspace layout (floats); e2 and emb alias (e2 dead before emb written)
  float* e1    = ws;                               // 16,777,216
  float* p2o   = ws + 16777216;                    //  4,194,304
  float* sh    = ws + 16777216 + 4194304;          // 16,777,216
  float* e2    = sh;
  float* emb   = sh;
  float* feats = sh + 16777216;                    //     65,536
  float* cn    = feats + 65536;                    //      2,048

  k_cnorm<<<P_ / 256, 256, 0, stream>>>(ce, cn);
  // e1: B=16 x 8 o-tiles(32) x 64 n-tiles = 8192 waves -> 1024 blocks
  k_proj<<<1024, 256, 0, stream>>>(p1, pw, e1, C1_, S1_, 0);
  // e2: 16 x 8 x 16 = 2048 waves -> 256 blocks
  k_proj<<<256, 256, 0, stream>>>(p2, pw, e2, C2_, S2_, C1_);
  k_pool2<<<(B_ * OUT_ * S2_) / 256, 256, 0, stream>>>(e2, p2o);
  k_emb<<<(B_ * OUT_ * S1_) / 256, 256, 0, stream>>>(e1, p2o, pb, emb);
  k_feats<<<(B_ * S1_) / 256, 256, 0, stream>>>(emb, feats);
  // 16 batches x 128 pixel-tiles(32) = 2048 waves -> 256 blocks
  k_dist<<<256, 256, 0, stream>>>(emb, ce, feats, cn, out);
}